// GATEncoderLayer_87076166959574
// MI455X (gfx1250) — compile-verified
//
#include <hip/hip_runtime.h>

// ---------------------------------------------------------------------------
// GATv2 x2 layers for MI455X (gfx1250, wave32).
// GEMMs use V_WMMA_F32_16X16X4_F32 (fp32 reference -> fp32 matrix cores).
// Edge phase: 2 passes (score+segmentMax, then exp+aggregate with deferred
// division by the softmax denominator), all traffic L2-resident (~41MB tables
// vs 192MB L2).
// ---------------------------------------------------------------------------

typedef float v2f __attribute__((ext_vector_type(2)));
typedef float v8f __attribute__((ext_vector_type(8)));

#define HC 256          // H*C
#define NEG_SLOPE 0.2f

// ---------------------------------------------------------------------------
// Dual GEMM: XL = A*Wl + bl, XR = A*Wr + br.   A:[Mx K], W:[K x 256].
// One wave -> one 16x16 tile of each output (A fragment shared).
// grid = (M/16, 256/128), block = 256 (8 waves). M,K multiples of 16/4.
// ---------------------------------------------------------------------------
__global__ void __launch_bounds__(256) gemm_dual_wmma(
    const float* __restrict__ A,
    const float* __restrict__ Wl, const float* __restrict__ bl,
    const float* __restrict__ Wr, const float* __restrict__ br,
    float* __restrict__ XL, float* __restrict__ XR, int K) {
  const int lane  = threadIdx.x & 31;
  const int wave  = threadIdx.x >> 5;
  const int mtile = blockIdx.x << 4;
  const int ntile = ((blockIdx.y << 3) + wave) << 4;
  const int idx16 = lane & 15;            // M index for A, N index for B
  const int kb    = (lane >> 4) << 1;     // lanes 16-31 hold K+2,K+3

  v8f cl = {};
  v8f cr = {};

  const float* arow = A  + (size_t)(mtile + idx16) * K;
  const float* wlc  = Wl + ntile + idx16;
  const float* wrc  = Wr + ntile + idx16;

  for (int k0 = 0; k0 < K; k0 += 4) {
    v2f a, b0, b1;
    a.x  = arow[k0 + kb];
    a.y  = arow[k0 + kb + 1];
    b0.x = wlc[(size_t)(k0 + kb) * HC];
    b0.y = wlc[(size_t)(k0 + kb + 1) * HC];
    b1.x = wrc[(size_t)(k0 + kb) * HC];
    b1.y = wrc[(size_t)(k0 + kb + 1) * HC];
    cl = __builtin_amdgcn_wmma_f32_16x16x4_f32(false, a, false, b0,
                                               (short)0, cl, false, false);
    cr = __builtin_amdgcn_wmma_f32_16x16x4_f32(false, a, false, b1,
                                               (short)0, cr, false, false);
  }

  // C/D layout: VGPR r -> row M = r + 8*(lane>=16), col N = lane&15.
  const int nc    = ntile + idx16;
  const int rbase = mtile + ((lane >> 4) << 3);
  const float bll = bl[nc];
  const float brr = br[nc];
#pragma unroll
  for (int r = 0; r < 8; ++r) {
    XL[(size_t)(rbase + r) * HC + nc] = cl[r] + bll;
    XR[(size_t)(rbase + r) * HC + nc] = cr[r] + brr;
  }
}

// ---------------------------------------------------------------------------
// Init per-layer accumulators: m = -inf, s = 0, agg = 0.
// ---------------------------------------------------------------------------
__global__ void init_kernel(float* __restrict__ m, float* __restrict__ s,
                            float* __restrict__ agg, int N) {
  const int i = blockIdx.x * blockDim.x + threadIdx.x;
  if (i < N * 4) { m[i] = -3.0e38f; s[i] = 0.0f; }
  if (i < N * HC) agg[i] = 0.0f;
}

// Sign-aware float atomic max via integer atomics (works for all finite f32).
__device__ inline void atomicMaxFloat(float* addr, float val) {
  if (val >= 0.0f)
    atomicMax((int*)addr, __float_as_int(val));
  else
    atomicMin((unsigned int*)addr, __float_as_uint(val));
}

// ---------------------------------------------------------------------------
// Edge pass 1: score[e][h] = sum_c att[h*64+c]*leakyrelu(xl[src]+xr[dst]),
// segment max into m[dst][h]. One wave per edge (8 channels/lane).
// Edges [0,E) from edge_index; [E, E+N) are self loops.
// ---------------------------------------------------------------------------
__global__ void __launch_bounds__(256) edge_score_kernel(
    const float* __restrict__ XL, const float* __restrict__ XR,
    const int* __restrict__ src, const int* __restrict__ dst,
    const float* __restrict__ att, float* __restrict__ score,
    float* __restrict__ m, int E, int N) {
  const int wave = (blockIdx.x * blockDim.x + threadIdx.x) >> 5;
  const int lane = threadIdx.x & 31;
  if (wave >= E + N) return;
  const int s_ = (wave < E) ? src[wave] : (wave - E);
  const int d_ = (wave < E) ? dst[wave] : (wave - E);
  const float* xl = XL + (size_t)s_ * HC;
  const float* xr = XR + (size_t)d_ * HC;

  float acc[4] = {0.f, 0.f, 0.f, 0.f};
#pragma unroll
  for (int j = 0; j < 8; ++j) {           // channel = lane + 32*j, head = j>>1
    const int ch = lane + (j << 5);
    float v = xl[ch] + xr[ch];
    v = (v > 0.0f) ? v : NEG_SLOPE * v;
    acc[j >> 1] += att[ch] * v;
  }
#pragma unroll
  for (int off = 16; off > 0; off >>= 1) {
#pragma unroll
    for (int h = 0; h < 4; ++h) acc[h] += __shfl_xor(acc[h], off, 32);
  }
  if (lane < 4) {
    score[(size_t)wave * 4 + lane] = acc[lane];
    atomicMaxFloat(&m[d_ * 4 + lane], acc[lane]);
  }
}

// ---------------------------------------------------------------------------
// Edge pass 2: p = exp(score - m[dst]); s[dst] += p; agg[dst] += p*xl[src].
// Division by s deferred to node epilogue.
// ---------------------------------------------------------------------------
__global__ void __launch_bounds__(256) edge_aggr_kernel(
    const float* __restrict__ XL, const int* __restrict__ src,
    const int* __restrict__ dst, const float* __restrict__ score,
    const float* __restrict__ m, float* __restrict__ s,
    float* __restrict__ agg, int E, int N) {
  const int wave = (blockIdx.x * blockDim.x + threadIdx.x) >> 5;
  const int lane = threadIdx.x & 31;
  if (wave >= E + N) return;
  const int s_ = (wave < E) ? src[wave] : (wave - E);
  const int d_ = (wave < E) ? dst[wave] : (wave - E);

  const float4 sc = *reinterpret_cast<const float4*>(score + (size_t)wave * 4);
  const float4 mm = *reinterpret_cast<const float4*>(m + (size_t)d_ * 4);
  float p[4];
  p[0] = __expf(sc.x - mm.x);
  p[1] = __expf(sc.y - mm.y);
  p[2] = __expf(sc.z - mm.z);
  p[3] = __expf(sc.w - mm.w);

  if (lane < 4) atomicAdd(&s[d_ * 4 + lane], p[lane]);

  const float* xl = XL + (size_t)s_ * HC;
  float* ag = agg + (size_t)d_ * HC;
#pragma unroll
  for (int j = 0; j < 8; ++j) {
    const int ch = lane + (j << 5);
    atomicAdd(&ag[ch], p[j >> 1] * xl[ch]);
  }
}

// ---------------------------------------------------------------------------
// Layer-1 epilogue: h1 = ELU(agg/s + bias).
// ---------------------------------------------------------------------------
__global__ void node_epilogue1(const float* __restrict__ agg,
                               const float* __restrict__ s,
                               const float* __restrict__ bias,
                               float* __restrict__ h1, int N) {
  const int i = blockIdx.x * blockDim.x + threadIdx.x;
  if (i >= N * HC) return;
  const int n = i >> 8, ch = i & 255;
  float v = agg[i] / s[n * 4 + (ch >> 6)] + bias[ch];
  h1[i] = (v > 0.0f) ? v : (__expf(v) - 1.0f);
}

// ---------------------------------------------------------------------------
// Layer-2 epilogue: h = agg/s + bias -> out0; log_softmax(h) -> out1.
// One 256-thread block per node.
// ---------------------------------------------------------------------------
__global__ void __launch_bounds__(256) node_epilogue2(
    const float* __restrict__ agg, const float* __restrict__ s,
    const float* __restrict__ bias, float* __restrict__ outh,
    float* __restrict__ outls, int N) {
  __shared__ float red[256];
  const int n = blockIdx.x, ch = threadIdx.x;
  const float v = agg[(size_t)n * HC + ch] / s[n * 4 + (ch >> 6)] + bias[ch];
  outh[(size_t)n * HC + ch] = v;

  red[ch] = v;
  __syncthreads();
#pragma unroll
  for (int o = 128; o > 0; o >>= 1) {
    if (ch < o) red[ch] = fmaxf(red[ch], red[ch + o]);
    __syncthreads();
  }
  const float mx = red[0];
  __syncthreads();
  red[ch] = __expf(v - mx);
  __syncthreads();
#pragma unroll
  for (int o = 128; o > 0; o >>= 1) {
    if (ch < o) red[ch] += red[ch + o];
    __syncthreads();
  }
  const float lse = mx + __logf(red[0]);
  outls[(size_t)n * HC + ch] = v - lse;
}

// ---------------------------------------------------------------------------
extern "C" void kernel_launch(void* const* d_in, const int* in_sizes, int n_in,
                              void* d_out, int out_size, void* d_ws,
                              size_t ws_size, hipStream_t stream) {
  const float* x     = (const float*)d_in[0];
  const int*   ei    = (const int*)d_in[1];
  const float* Wl1   = (const float*)d_in[2];
  const float* bl1   = (const float*)d_in[3];
  const float* Wr1   = (const float*)d_in[4];
  const float* br1   = (const float*)d_in[5];
  const float* att1  = (const float*)d_in[6];
  const float* bias1 = (const float*)d_in[7];
  const float* Wl2   = (const float*)d_in[8];
  const float* bl2   = (const float*)d_in[9];
  const float* Wr2   = (const float*)d_in[10];
  const float* br2   = (const float*)d_in[11];
  const float* att2  = (const float*)d_in[12];
  const float* bias2 = (const float*)d_in[13];

  const int Fin  = in_sizes[2] / HC;    // 64
  const int N    = in_sizes[0] / Fin;   // 20000
  const int E    = in_sizes[1] / 2;     // 500000
  const int Etot = E + N;               // + self loops
  const int* src = ei;
  const int* dst = ei + E;

  // Workspace layout (floats): XL, XR, H1, AGG [N*256 each], SCORE [Etot*4],
  // M [N*4], S [N*4].  ~91 MB total.
  float* ws    = (float*)d_ws;
  float* XL    = ws;
  float* XR    = XL + (size_t)N * HC;
  float* H1    = XR + (size_t)N * HC;
  float* AGG   = H1 + (size_t)N * HC;
  float* SCORE = AGG + (size_t)N * HC;
  float* Mx    = SCORE + (size_t)Etot * 4;
  float* Ss    = Mx + (size_t)N * 4;

  float* outh  = (float*)d_out;
  float* outls = outh + (size_t)N * HC;

  const dim3 gblk(256);
  const dim3 ggrid(N / 16, HC / 128);          // 1250 x 2, 8 waves/block
  const int nblk  = (N * HC + 255) / 256;      // node-granularity launches
  const int eblk  = (Etot + 7) / 8;            // one wave per edge

  // ---- Layer 1 ----
  gemm_dual_wmma<<<ggrid, gblk, 0, stream>>>(x, Wl1, bl1, Wr1, br1, XL, XR, Fin);
  init_kernel<<<nblk, 256, 0, stream>>>(Mx, Ss, AGG, N);
  edge_score_kernel<<<eblk, 256, 0, stream>>>(XL, XR, src, dst, att1, SCORE, Mx, E, N);
  edge_aggr_kernel<<<eblk, 256, 0, stream>>>(XL, src, dst, SCORE, Mx, Ss, AGG, E, N);
  node_epilogue1<<<nblk, 256, 0, stream>>>(AGG, Ss, bias1, H1, N);

  // ---- Layer 2 ----
  gemm_dual_wmma<<<ggrid, gblk, 0, stream>>>(H1, Wl2, bl2, Wr2, br2, XL, XR, HC);
  init_kernel<<<nblk, 256, 0, stream>>>(Mx, Ss, AGG, N);
  edge_score_kernel<<<eblk, 256, 0, stream>>>(XL, XR, src, dst, att2, SCORE, Mx, E, N);
  edge_aggr_kernel<<<eblk, 256, 0, stream>>>(XL, src, dst, SCORE, Mx, Ss, AGG, E, N);
  node_epilogue2<<<N, 256, 0, stream>>>(AGG, Ss, bias2, outh, outls, N);
}